// CIR_Y_56066503082266
// MI455X (gfx1250) — compile-verified
//
#include <hip/hip_runtime.h>
#include <hip/hip_bf16.h>
#include <stdint.h>

// ---------------------------------------------------------------------------
// CIR Euler-Maruyama scan, MI455X (gfx1250).
// Memory-bound streaming kernel (~272MB @ 23.3TB/s => ~12us floor).
// - 2 independent rows per lane (ILP to hide sqrt/fma dependent-chain latency
//   under low occupancy: only 16384 serial chains exist).
// - W staged global->LDS with CDNA5 async-to-LDS (ASYNCcnt), double buffered,
//   fully coalesced (each async b128 instruction covers 4 whole cachelines).
// - Output transposed through LDS so global stores are full-cacheline b128,
//   emitted non-temporal (write-once stream, keep L2 for W).
// - Wave-private tiles: no workgroup barriers (per-wave DS ordering suffices).
// ---------------------------------------------------------------------------

#define B_TOTAL   16384
#define S_LEN     2048
#define L_DIM     64
#define TS        32                     // columns (time steps) per stage
#define NSTAGE    (S_LEN / TS)           // 64
#define WAVE_SZ   32
#define WAVES_PB  2
#define BLOCK_T   (WAVE_SZ * WAVES_PB)   // 64 threads
#define ROWS_PW   64                     // 2 rows per lane
#define ROWS_PB   (ROWS_PW * WAVES_PB)   // 128 rows per block
#define GRID_B    (B_TOTAL / ROWS_PB)    // 128 blocks

typedef float v4f __attribute__((ext_vector_type(4)));

__device__ __forceinline__ float cir_step(float v, float w, float kdt,
                                          float tmu, float swl) {
  // v_new = v + kdt*(tmu - v) + swl*sqrt(max(v,0))*w
  // (sqrt(clip(v*DT,0)) == sqrt(DT)*sqrt(clip(v,0)); swl = sigma*sqrt(DT))
  float sw = swl * w;                     // off the critical path
#if __has_builtin(__builtin_amdgcn_sqrtf)
  float s = __builtin_amdgcn_sqrtf(__builtin_fmaxf(v, 0.0f));
#else
  float s = __builtin_sqrtf(__builtin_fmaxf(v, 0.0f));
#endif
  return __builtin_fmaf(sw, s, __builtin_fmaf(kdt, tmu - v, v));
}

__global__ __launch_bounds__(BLOCK_T) void cir_scan_kernel(
    const float* __restrict__ x, const float* __restrict__ W,
    const float* __restrict__ kap, const float* __restrict__ muv,
    const float* __restrict__ sig, float* __restrict__ out) {

  // Per-wave tiles. win[wave][buf][sub][chunk][row]: element = W(row, chunk*4..+3)
  // of the sub-th 32-row group. wout is the same shape (+row pad) for outputs.
  __shared__ v4f win [WAVES_PB][2][2][8][32];   // 32 KB
  __shared__ v4f wout[WAVES_PB][2][8][33];      // ~16.9 KB

  const int lane = threadIdx.x & (WAVE_SZ - 1);
  const int wv   = threadIdx.x >> 5;
  const int b0w  = blockIdx.x * ROWS_PB + wv * ROWS_PW;  // first of 64 rows

  // -------- per-lane address precompute for async staging ------------------
  const uint64_t wbase   = (uint64_t)W;
  // async instr i, lane L: row = i*4 + L/8, 16B chunk = L%8  -> 4 full lines.
  const uint32_t laneoff = (uint32_t)((lane >> 3) * (S_LEN * 4) + (lane & 7) * 16);
  const uint32_t row0off = (uint32_t)b0w * (uint32_t)(S_LEN * 4);
  const uint32_t ldl     = (uint32_t)((lane & 7) * 512 + (lane >> 3) * 16);

  auto issue_stage = [&](int stage, int buf) {
    const uint32_t stb = (uint32_t)stage * (uint32_t)(TS * 4);  // 128B/stage
#pragma unroll
    for (int sub = 0; sub < 2; ++sub) {
      const uint32_t g0 = row0off + (uint32_t)sub * (32u * S_LEN * 4u) + laneoff + stb;
      const uint32_t d0 =
          (uint32_t)(uintptr_t)(&win[wv][buf][sub][0][0]) + ldl;
#pragma unroll
      for (int i = 0; i < 8; ++i) {
        const uint32_t voff = g0 + (uint32_t)i * (4u * S_LEN * 4u);
        const uint32_t dst  = d0 + (uint32_t)i * 64u;
        // GVS: mem = SGPR64 + VGPR32 ; VDST = per-lane LDS byte address.
        asm volatile("global_load_async_to_lds_b128 %0, %1, %2"
                     :: "v"(dst), "v"(voff), "s"(wbase)
                     : "memory");
      }
    }
  };

  // -------- per-row constants (xmean over 64 samples) ----------------------
  const float kappa = kap[0], mu = muv[0], sigma = sig[0];
  const float DT = 1.0f / (float)S_LEN;

  float xm[2];
#pragma unroll
  for (int sub = 0; sub < 2; ++sub) {
    const int r = b0w + sub * 32 + lane;
    const v4f* xr = (const v4f*)(x + (size_t)r * L_DIM);
    float s = 0.0f;
#pragma unroll
    for (int j = 0; j < L_DIM / 4; ++j) {
      v4f t = xr[j];
      s += (t.x + t.y) + (t.z + t.w);
    }
    xm[sub] = s * (1.0f / (float)L_DIM);
  }

  const float kdt  = kappa * DT;
  const float swl  = sigma * __builtin_sqrtf(DT);
  const float tmu0 = mu + xm[0], tmu1 = mu + xm[1];
  const float ox0  = 0.5f * xm[0], ox1 = 0.5f * xm[1];
  float v0 = 0.04f, v1 = 0.04f;

  // -------- main pipeline --------------------------------------------------
  issue_stage(0, 0);

  for (int st = 0; st < NSTAGE; ++st) {
    const int buf = st & 1;
    if (st + 1 < NSTAGE) {
      issue_stage(st + 1, buf ^ 1);
      // 16 loads just issued; in-order completion => <=16 means stage st done.
      asm volatile("s_wait_asynccnt 16" ::: "memory");
    } else {
      asm volatile("s_wait_asynccnt 0" ::: "memory");
    }

    // 32 steps for both rows, 2-way ILP (independent chains interleaved).
#pragma unroll
    for (int c4 = 0; c4 < 8; ++c4) {
      v4f wa = win[wv][buf][0][c4][lane];
      v4f wb = win[wv][buf][1][c4][lane];
      v4f oa, ob;
      v0 = cir_step(v0, wa.x, kdt, tmu0, swl);
      v1 = cir_step(v1, wb.x, kdt, tmu1, swl);
      oa.x = __builtin_fmaf(0.5f, v0, ox0);
      ob.x = __builtin_fmaf(0.5f, v1, ox1);
      v0 = cir_step(v0, wa.y, kdt, tmu0, swl);
      v1 = cir_step(v1, wb.y, kdt, tmu1, swl);
      oa.y = __builtin_fmaf(0.5f, v0, ox0);
      ob.y = __builtin_fmaf(0.5f, v1, ox1);
      v0 = cir_step(v0, wa.z, kdt, tmu0, swl);
      v1 = cir_step(v1, wb.z, kdt, tmu1, swl);
      oa.z = __builtin_fmaf(0.5f, v0, ox0);
      ob.z = __builtin_fmaf(0.5f, v1, ox1);
      v0 = cir_step(v0, wa.w, kdt, tmu0, swl);
      v1 = cir_step(v1, wb.w, kdt, tmu1, swl);
      oa.w = __builtin_fmaf(0.5f, v0, ox0);
      ob.w = __builtin_fmaf(0.5f, v1, ox1);
      wout[wv][0][c4][lane] = oa;
      wout[wv][1][c4][lane] = ob;
    }

    __builtin_amdgcn_wave_barrier();   // wave-synchronous LDS handoff

    // Transposed drain: each b128 NT store writes whole 128B lines.
    const int s0 = st * TS;
#pragma unroll
    for (int sub = 0; sub < 2; ++sub) {
#pragma unroll
      for (int i = 0; i < 8; ++i) {
        const int row = i * 4 + (lane >> 3);
        v4f d = wout[wv][sub][lane & 7][row];
        v4f* op = (v4f*)(out + (size_t)(b0w + sub * 32 + row) * S_LEN + s0)
                  + (lane & 7);
        __builtin_nontemporal_store(d, op);
      }
    }
    __builtin_amdgcn_wave_barrier();
  }
}

extern "C" void kernel_launch(void* const* d_in, const int* in_sizes, int n_in,
                              void* d_out, int out_size, void* d_ws, size_t ws_size,
                              hipStream_t stream) {
  (void)in_sizes; (void)n_in; (void)out_size; (void)d_ws; (void)ws_size;
  const float* x   = (const float*)d_in[0];
  const float* W   = (const float*)d_in[1];
  const float* kap = (const float*)d_in[2];
  const float* mu  = (const float*)d_in[3];
  const float* sig = (const float*)d_in[4];
  float* out = (float*)d_out;
  cir_scan_kernel<<<dim3(GRID_B), dim3(BLOCK_T), 0, stream>>>(x, W, kap, mu, sig, out);
}